// FM_14336600834857
// MI455X (gfx1250) — compile-verified
//
#include <hip/hip_runtime.h>

typedef __attribute__((ext_vector_type(2))) float v2f;
typedef __attribute__((ext_vector_type(8))) float v8f;

#define F_DIM 200
#define E_DIM 64
#define LDS_PITCH 202              // pad 200 -> 202 dwords: keeps 8B align, spreads banks
#define WAVES_PER_BLOCK 8
#define ROWS_PER_WAVE 16
#define ROWS_PER_BLOCK (WAVES_PER_BLOCK * ROWS_PER_WAVE)   // 128

// out[b] = sum_e ( (features@W)[b,e]^2 ) - sum_f features[b,f]^2 * (sum_e W[f,e]^2)
__global__ __launch_bounds__(256)
void fm_second_order_kernel(const float* __restrict__ features,
                            const float* __restrict__ W,
                            float* __restrict__ out, int Brows) {
    __shared__ float sWt[E_DIM * LDS_PITCH];   // W transposed: sWt[e*PITCH + f]
    __shared__ float sW2[F_DIM];               // w2sum[f] = sum_e W[f,e]^2

    const int tid  = threadIdx.x;
    const int lane = tid & 31;
    const int wave = tid >> 5;

    // Stage W transposed into LDS (coalesced global reads of row-major W[f][e]).
    for (int i = tid; i < F_DIM * E_DIM; i += 256) {
        int f = i >> 6;          // / E_DIM
        int e = i & (E_DIM - 1);
        sWt[e * LDS_PITCH + f] = W[i];
    }
    __syncthreads();
    // w2sum per feature.
    if (tid < F_DIM) {
        float s = 0.f;
        #pragma unroll
        for (int e = 0; e < E_DIM; ++e) {
            float w = sWt[e * LDS_PITCH + tid];
            s = fmaf(w, w, s);
        }
        sW2[tid] = s;
    }
    __syncthreads();

    const int base_row = blockIdx.x * ROWS_PER_BLOCK + wave * ROWS_PER_WAVE;
    if (base_row >= Brows) return;             // uniform per wave; EXEC stays all-ones

    const int mrow = lane & 15;                // A-frag row for this lane
    const int half = lane >> 4;                // 0 -> K pair {0,1}, 1 -> {2,3}
    const int n0   = lane & 15;                // B-frag column within 16-wide tile

    const float* arow = features + (size_t)(base_row + mrow) * F_DIM;

    v8f acc0 = {}, acc1 = {}, acc2 = {}, acc3 = {};
    float sqp = 0.f;                           // partial of sum_f a^2 * w2sum[f]

    for (int k = 0; k < F_DIM; k += 4) {
        const int kk = k + 2 * half;
        // A fragment (f32 16x4 layout): VGPR0 = K even, VGPR1 = K odd, halves split K.
        v2f a = *reinterpret_cast<const v2f*>(arow + kk);
        // fold the squared-term mat-vec into the same loads
        sqp = fmaf(a.x * a.x, sW2[kk],     sqp);
        sqp = fmaf(a.y * a.y, sW2[kk + 1], sqp);
        // B fragments (4x16) for the four 16-wide slices of E=64, from transposed LDS
        v2f b0 = *reinterpret_cast<const v2f*>(&sWt[(n0 +  0) * LDS_PITCH + kk]);
        v2f b1 = *reinterpret_cast<const v2f*>(&sWt[(n0 + 16) * LDS_PITCH + kk]);
        v2f b2 = *reinterpret_cast<const v2f*>(&sWt[(n0 + 32) * LDS_PITCH + kk]);
        v2f b3 = *reinterpret_cast<const v2f*>(&sWt[(n0 + 48) * LDS_PITCH + kk]);
        acc0 = __builtin_amdgcn_wmma_f32_16x16x4_f32(false, a, false, b0, (short)0, acc0, false, false);
        acc1 = __builtin_amdgcn_wmma_f32_16x16x4_f32(false, a, false, b1, (short)0, acc1, false, false);
        acc2 = __builtin_amdgcn_wmma_f32_16x16x4_f32(false, a, false, b2, (short)0, acc2, false, false);
        acc3 = __builtin_amdgcn_wmma_f32_16x16x4_f32(false, a, false, b3, (short)0, acc3, false, false);
    }

    // Per-lane sum of squares across its N columns for the 8 M rows it holds.
    // C layout: VGPR r, lanes 0-15 -> (M=r, N=lane); lanes 16-31 -> (M=r+8, N=lane-16).
    float s[8];
    #pragma unroll
    for (int r = 0; r < 8; ++r) {
        float c0 = acc0[r], c1 = acc1[r], c2 = acc2[r], c3 = acc3[r];
        s[r] = c0 * c0 + c1 * c1 + c2 * c2 + c3 * c3;
    }
    // Reduce across the 16 lanes of each half-wave (xor masks <= 8 never cross halves).
    #pragma unroll
    for (int m = 1; m <= 8; m <<= 1) {
        #pragma unroll
        for (int r = 0; r < 8; ++r) s[r] += __shfl_xor(s[r], m, 32);
    }
    // Full squared-term for row mrow (low lanes hold K%4 in {0,1}, high lanes {2,3}).
    float sqfull = sqp + __shfl_xor(sqp, 16, 32);

    // Writers: lanes 0-7 -> rows 0-7 (from low-half s[]); lanes 16-23 -> rows 8-15.
    int myrow = (lane & 7) + ((lane >> 4) << 3);
    float sqr = __shfl(sqfull, myrow, 32);     // sq[row] lives at lane == row (row < 16)
    if ((lane & 15) < 8) {
        out[base_row + myrow] = s[lane & 7] - sqr;
    }
}

extern "C" void kernel_launch(void* const* d_in, const int* in_sizes, int n_in,
                              void* d_out, int out_size, void* d_ws, size_t ws_size,
                              hipStream_t stream) {
    const float* features = (const float*)d_in[0];   // [B, 200] f32
    const float* W        = (const float*)d_in[1];   // [200, 64] f32
    float* out            = (float*)d_out;           // [B, 1] f32
    (void)d_ws; (void)ws_size; (void)n_in; (void)out_size;

    int B = in_sizes[0] / F_DIM;                     // 16384
    int blocks = (B + ROWS_PER_BLOCK - 1) / ROWS_PER_BLOCK;
    fm_second_order_kernel<<<blocks, 256, 0, stream>>>(features, W, out, B);
}